// VectorQuantizer_14851996909601
// MI455X (gfx1250) — compile-verified
//
#include <hip/hip_runtime.h>

// CDNA5 / gfx1250, wave32. Exact-precision path: v_wmma_f32_16x16x4_f32.
typedef float v2f __attribute__((ext_vector_type(2)));
typedef float v8f __attribute__((ext_vector_type(8)));

#define DIM              64
#define NCODES           1024
#define CHUNK            128                    // codebook rows staged in LDS at a time
#define NCHUNKS          (NCODES / CHUNK)       // 8
#define TILES_PER_CHUNK  (CHUNK / 16)           // 8
#define WAVES_PER_BLOCK  8
#define ROWS_PER_BLOCK   (16 * WAVES_PER_BLOCK) // 128
#define BLOCK_THREADS    256

__global__ __launch_bounds__(BLOCK_THREADS)
void vq_argmin_gather_kernel(const float* __restrict__ x,
                             const float* __restrict__ W,
                             float* __restrict__ outA,   // quantized_with_grad
                             float* __restrict__ outB,   // quantized
                             int*   __restrict__ outIdx) // indices
{
    __shared__ float sW[CHUNK * DIM];   // 32 KB: chunk of codebook
    __shared__ float sWsq[CHUNK];       // 0.5 KB: per-code ||w||^2

    const int tid  = threadIdx.x;
    const int wave = tid >> 5;
    const int lane = tid & 31;
    const int half = lane >> 4;   // 0: lanes 0-15, 1: lanes 16-31
    const int l16  = lane & 15;

    const int m0 = blockIdx.x * ROWS_PER_BLOCK + wave * 16;

    // ---- Preload A fragments: 16 rows x 64 dims of x, kept in VGPRs for the
    // whole kernel. ISA 16x4 f32 A layout: lane holds row m=l16, VGPR pair kb
    // covers K = 4*kb + 2*half {+0,+1}  ->  contiguous float2 loads.
    v2f a[16];
    {
        const float* xr = x + (size_t)(m0 + l16) * DIM + half * 2;
        #pragma unroll
        for (int kb = 0; kb < 16; ++kb)
            a[kb] = *(const v2f*)(xr + kb * 4);
    }

    // Running argmin of (w_sq[n] - 2*dot) per lane; f_sq[m] is row-constant
    // and cannot change the argmin, so it is dropped.
    float bestd[8];
    int   bestn[8];
    #pragma unroll
    for (int r = 0; r < 8; ++r) { bestd[r] = 3.0e38f; bestn[r] = 0; }

    for (int ch = 0; ch < NCHUNKS; ++ch) {
        __syncthreads();   // previous chunk's LDS consumers done
        // Stage CHUNK codebook rows into LDS (coalesced float4 copies).
        {
            const float4* gsrc = (const float4*)(W + (size_t)ch * CHUNK * DIM);
            float4* ldst = (float4*)sW;
            #pragma unroll
            for (int i = 0; i < (CHUNK * DIM / 4) / BLOCK_THREADS; ++i)
                ldst[tid + i * BLOCK_THREADS] = gsrc[tid + i * BLOCK_THREADS];
        }
        __syncthreads();
        // ||w||^2 for this chunk (one code per thread, threads 0..127).
        if (tid < CHUNK) {
            const float* wr = sW + tid * DIM;
            float s = 0.f;
            #pragma unroll
            for (int k = 0; k < DIM; ++k) s += wr[k] * wr[k];
            sWsq[tid] = s;
        }
        __syncthreads();

        // ---- 8 code tiles of 16; each tile = 16 WMMA f32 16x16x4 steps.
        #pragma unroll 1
        for (int nt = 0; nt < TILES_PER_CHUNK; ++nt) {
            // B 4x16 layout mirrors A: lane holds code n = nt*16 + l16,
            // K = 4*kb + 2*half {+0,+1}  ->  contiguous float2 LDS loads.
            const float* wb = sW + (size_t)(nt * 16 + l16) * DIM + half * 2;

            // Hoist all 16 B fragments first (8x ds_load_2addr_b64, one
            // dscnt wait) so the 16-WMMA chain runs without LDS stalls.
            v2f b[16];
            #pragma unroll
            for (int kb = 0; kb < 16; ++kb)
                b[kb] = *(const v2f*)(wb + kb * 4);

            v8f c0 = {};
            v8f c1 = {};   // two accumulator chains to cover XDL latency
            #pragma unroll
            for (int kb = 0; kb < 16; kb += 2) {
                c0 = __builtin_amdgcn_wmma_f32_16x16x4_f32(
                        false, a[kb],     false, b[kb],     (short)0, c0, false, false);
                c1 = __builtin_amdgcn_wmma_f32_16x16x4_f32(
                        false, a[kb + 1], false, b[kb + 1], (short)0, c1, false, false);
            }

            const int   n   = ch * CHUNK + nt * 16 + l16;  // this lane's column
            const float wsq = sWsq[nt * 16 + l16];
            // C layout: VGPR r -> row (r + 8*half), column l16.
            #pragma unroll
            for (int r = 0; r < 8; ++r) {
                float dist = wsq - 2.0f * (c0[r] + c1[r]);
                bool  upd  = dist < bestd[r];   // strict '<': keeps first min
                bestd[r] = upd ? dist : bestd[r];
                bestn[r] = upd ? n    : bestn[r];
            }
        }
    }

    // ---- Cross-lane argmin over the 16 lanes holding each row, then gather
    // the winning codebook row and write both outputs + index.
    #pragma unroll
    for (int r = 0; r < 8; ++r) {
        float d = bestd[r];
        int   n = bestn[r];
        #pragma unroll
        for (int off = 1; off < 16; off <<= 1) {   // stays within the half-wave
            float od = __shfl_xor(d, off, 32);
            int   on = __shfl_xor(n, off, 32);
            bool take = (od < d) || (od == d && on < n);  // jnp.argmin tie-break
            d = take ? od : d;
            n = take ? on : n;
        }
        const int row = m0 + r + 8 * half;
        // 16 lanes x float4 = the full 64-float code row, coalesced 256B.
        const float4 qv = *(const float4*)(W + (size_t)n * DIM + l16 * 4);
        *((float4*)(outA + (size_t)row * DIM) + l16) = qv;
        *((float4*)(outB + (size_t)row * DIM) + l16) = qv;
        if (l16 == 0) outIdx[row] = n;
    }
}

extern "C" void kernel_launch(void* const* d_in, const int* in_sizes, int n_in,
                              void* d_out, int out_size, void* d_ws, size_t ws_size,
                              hipStream_t stream) {
    const float* x = (const float*)d_in[0];   // [8,128,128,64] f32
    const float* W = (const float*)d_in[1];   // [1024,64] f32

    const int nrows = in_sizes[0] / DIM;      // 131072

    // Tuple outputs concatenated flat: qwg [N*64] f32, q [N*64] f32, idx [N] i32.
    float* outA   = (float*)d_out;
    float* outB   = outA + (size_t)nrows * DIM;
    int*   outIdx = (int*)(outB + (size_t)nrows * DIM);

    dim3 grid(nrows / ROWS_PER_BLOCK);        // 1024 workgroups
    vq_argmin_gather_kernel<<<grid, BLOCK_THREADS, 0, stream>>>(
        x, W, outA, outB, outIdx);
}